// SegmentAwarePool_69148973465778
// MI455X (gfx1250) — compile-verified
//
#include <hip/hip_runtime.h>
#include <hip/hip_bf16.h>

typedef __attribute__((ext_vector_type(2))) float v2f;
typedef __attribute__((ext_vector_type(8))) float v8f;

#define BATCH 256
#define SEQ   512
#define DIM   768

// ---------------------------------------------------------------------------
// Phase 1: per-batch segment metadata.
// One block (256 threads) per batch row. Collect SEP positions into a 512-bit
// shared bitmask, reduce attention_mask, then thread 0 extracts pos1..pos3.
// Meta layout (8 ints per batch in d_ws):
//   [0]=title_lo [1]=title_hi [2]=lead_lo [3]=lead_hi
//   [4]=bits(inv_tn) [5]=bits(inv_ln) [6]=use_cls_title [7]=use_cls_lead
// ---------------------------------------------------------------------------
__global__ __launch_bounds__(256) void sap_meta_kernel(
    const int* __restrict__ ids, const int* __restrict__ amask,
    const int* __restrict__ sep_tok, int* __restrict__ meta) {
  const int b = blockIdx.x;
  const int t = threadIdx.x;

  __shared__ unsigned sepbits[SEQ / 32];
  __shared__ int attn;
  if (t < SEQ / 32) sepbits[t] = 0u;
  if (t == 0) attn = 0;
  __syncthreads();

  const int sep_id = sep_tok[0];
  int local_attn = 0;
  #pragma unroll
  for (int s = t; s < SEQ; s += 256) {
    if (ids[b * SEQ + s] == sep_id)
      atomicOr(&sepbits[s >> 5], 1u << (s & 31));
    local_attn += amask[b * SEQ + s];
  }
  if (local_attn) atomicAdd(&attn, local_attn);
  __syncthreads();

  if (t == 0) {
    int pos[3] = {SEQ, SEQ, SEQ};
    int nsep = 0;
    for (int w = 0; w < SEQ / 32; ++w) {
      unsigned bits = sepbits[w];
      while (bits) {
        int bit = __ffs(bits) - 1;
        if (nsep < 3) pos[nsep] = w * 32 + bit;
        nsep++;
        bits &= bits - 1;
      }
    }
    const int t_lo = 1;
    const int t_hi = pos[0];                       // idx in [1, pos1)
    const int tn   = max(t_hi - t_lo, 0);
    const int l_lo = pos[1] + 1;                   // idx in [pos2+1, lead_end)
    const int l_hi = (nsep >= 3) ? pos[2] : attn;
    const int ln   = max(l_hi - l_lo, 0);
    const bool valid = (nsep >= 2);

    int* m = meta + b * 8;
    m[0] = t_lo;
    m[1] = t_hi;
    m[2] = l_lo;
    m[3] = l_hi;
    m[4] = __float_as_int(1.0f / (float)max(tn, 1));
    m[5] = __float_as_int(1.0f / (float)max(ln, 1));
    m[6] = (valid && tn > 0) ? 0 : 1;
    m[7] = (valid && ln > 0) ? 0 : 1;
  }
}

// ---------------------------------------------------------------------------
// Phase 2: masked-mean pooling via V_WMMA_F32_16X16X4_F32.
//   D = A(16x4) * B(4x16) + C
//   A row 0 = title mask over 4 seq positions, row 1 = lead mask, rows 2..15 = 0
//   B       = hidden[b, s:s+4, d0:d0+16]
// A layout (32-bit 16x4):  lanes 0-15: M=lane,    VGPR0=K0, VGPR1=K1
//                          lanes 16-31: M=lane-16, VGPR0=K2, VGPR1=K3
// B layout (32-bit 4x16):  lanes 0-15: N=lane, K = vgpr;  lanes 16-31: K = vgpr+2
// C layout: VGPR i, lanes 0-15 -> row M=i  =>  c[0]=title sums, c[1]=lead sums.
//
// Branch-free A fragment: each lane owns one span [lo, lo+span) (empty for
// rows >= 2); per K-step the mask is (unsigned)(k - lo) < span, with k - lo
// kept as a running register. No EXEC manipulation inside the loop.
// One wave per (batch, 16-col tile); 4 waves per block.
// ---------------------------------------------------------------------------
__global__ __launch_bounds__(128) void sap_pool_kernel(
    const float* __restrict__ h, const int* __restrict__ meta,
    float* __restrict__ out) {
  const int b    = blockIdx.y;
  const int lane = threadIdx.x & 31;
  const int wave = threadIdx.x >> 5;
  const int tile = blockIdx.x * 4 + wave;        // 0..47
  const int d0   = tile * 16;

  const int* m = meta + b * 8;
  const int   t_lo = m[0], t_hi = m[1], l_lo = m[2], l_hi = m[3];
  const float inv_tn = __int_as_float(m[4]);
  const float inv_ln = __int_as_float(m[5]);
  const int   use_cls_t = m[6], use_cls_l = m[7];

  const int mrow  = lane & 15;        // A: row M / B: column N
  const int khalf = (lane >> 4) * 2;  // upper half of wave holds K+2

  // Per-lane span owned by this A row (empty for rows 2..15).
  const int      lo_sel = (mrow == 0) ? t_lo : l_lo;
  const int      hi_sel = (mrow == 0) ? t_hi : l_hi;
  const unsigned span   = (mrow <= 1) ? (unsigned)max(hi_sel - lo_sel, 0) : 0u;

  const float* __restrict__ hb = h + (size_t)b * SEQ * DIM;
  const float* __restrict__ hp = hb + (size_t)khalf * DIM + d0 + mrow;

  // Running k - lo for the two A elements this lane holds (K and K+1).
  int r0 = khalf - lo_sel;
  int r1 = r0 + 1;

  v8f c = {};
  #pragma unroll 8
  for (int s = 0; s < SEQ; s += 4) {
    v2f a;
    a.x = ((unsigned)r0 < span) ? 1.0f : 0.0f;
    a.y = ((unsigned)r1 < span) ? 1.0f : 0.0f;

    // B fragment: hidden rows k0/k1 at column d0+mrow (64B coalesced runs)
    v2f bm;
    bm.x = hp[(size_t)s * DIM];
    bm.y = hp[(size_t)s * DIM + DIM];

    c = __builtin_amdgcn_wmma_f32_16x16x4_f32(
        /*neg_a=*/false, a, /*neg_b=*/false, bm,
        /*c_mod=*/(short)0, c, /*reuse_a=*/false, /*reuse_b=*/false);

    r0 += 4;
    r1 += 4;
  }

  if (lane < 16) {
    const int d = d0 + lane;
    const float cls  = hb[d];  // hidden[b, 0, d]
    const float tout = use_cls_t ? cls : c[0] * inv_tn;
    const float lout = use_cls_l ? cls : c[1] * inv_ln;
    out[(size_t)b * DIM + d] = tout;
    out[(size_t)BATCH * DIM + (size_t)b * DIM + d] = lout;
  }
}

// ---------------------------------------------------------------------------
extern "C" void kernel_launch(void* const* d_in, const int* in_sizes, int n_in,
                              void* d_out, int out_size, void* d_ws, size_t ws_size,
                              hipStream_t stream) {
  const float* hidden  = (const float*)d_in[0];
  const int*   ids     = (const int*)d_in[1];
  const int*   amask   = (const int*)d_in[2];
  const int*   sep_tok = (const int*)d_in[3];
  int*   meta = (int*)d_ws;              // 256 * 8 ints = 8 KB
  float* out  = (float*)d_out;

  hipLaunchKernelGGL(sap_meta_kernel, dim3(BATCH), dim3(256), 0, stream,
                     ids, amask, sep_tok, meta);
  hipLaunchKernelGGL(sap_pool_kernel, dim3(DIM / 64, BATCH), dim3(128), 0, stream,
                     hidden, meta, out);
}